// HierarchicalMultiLabelLoss_37306085933524
// MI455X (gfx1250) — compile-verified
//
#include <hip/hip_runtime.h>

typedef __attribute__((ext_vector_type(2))) float v2f;
typedef __attribute__((ext_vector_type(8))) float v8f;

#define NB    1024          // blocks for main kernel (ws usage: NB*8*4 = 32 KB)
#define BLK   256           // 8 waves per block (wave32)
#define WAVES (BLK / 32)

// labels are exactly 0.0 or 1.0, so BCE collapses to a single log:
//   -(y*log(p) + (1-y)*log1p(-p)) == -log(y ? p : 1-p)
__device__ __forceinline__ float bce01(float p, float y) {
    return -__logf(y > 0.5f ? p : 1.0f - p);
}

__global__ __launch_bounds__(BLK) void hml_main(
    const float* __restrict__ l1p,  const float* __restrict__ l2pcc, const float* __restrict__ l2pur,
    const float* __restrict__ p3a,  const float* __restrict__ p3b,   const float* __restrict__ p3c,
    const float* __restrict__ p3d,  const float* __restrict__ p3e,
    const float* __restrict__ l1y,  const float* __restrict__ l2ycc, const float* __restrict__ l2yur,
    const float* __restrict__ y3a,  const float* __restrict__ y3b,   const float* __restrict__ y3c,
    const float* __restrict__ y3d,  const float* __restrict__ y3e,
    float* __restrict__ ws, int n)
{
    const int lane  = threadIdx.x & 31;
    const int wave  = (blockIdx.x * BLK + threadIdx.x) >> 5;
    const int nwave = gridDim.x * WAVES;

    // 7 accumulators: l1 sum, l2cc sum, l2urw sum, cntCC, cntURW, l3 sum, l3 cnt
    float a0 = 0.f, a1 = 0.f, a2 = 0.f, a3 = 0.f, a4 = 0.f, a5 = 0.f, a6 = 0.f;

    for (int i = wave; i < n; i += nwave) {
        const int i2 = i * 2, i3 = i * 3, ib = i * 32 + lane;

        // wave-uniform label loads (broadcast; served once from cache)
        const float y1a = l1y[i2],     y1b = l1y[i2 + 1];
        const float c0  = l2ycc[i3],   c1  = l2ycc[i3 + 1], c2 = l2ycc[i3 + 2];
        const float u0  = l2yur[i2],   u1  = l2yur[i2 + 1];

        const bool actCC = y1a > 0.5f;
        const bool actUR = y1b > 0.5f;
        const float m0 = (actCC && c0 != 0.f) ? 1.f : 0.f;
        const float m1 = (actCC && c1 != 0.f) ? 1.f : 0.f;
        const float m2 = (actCC && c2 != 0.f) ? 1.f : 0.f;
        const float m3 = (actUR && u0 != 0.f) ? 1.f : 0.f;
        const float m4 = (actUR && u1 != 0.f) ? 1.f : 0.f;

        // level-3: lane == class; masks are wave-uniform -> inactive rows are
        // skipped entirely (no HBM traffic for ~75% of the dominant data)
        if (m0 != 0.f) a5 += bce01(__builtin_nontemporal_load(p3a + ib),
                                   __builtin_nontemporal_load(y3a + ib));
        if (m1 != 0.f) a5 += bce01(__builtin_nontemporal_load(p3b + ib),
                                   __builtin_nontemporal_load(y3b + ib));
        if (m2 != 0.f) a5 += bce01(__builtin_nontemporal_load(p3c + ib),
                                   __builtin_nontemporal_load(y3c + ib));
        if (m3 != 0.f) a5 += bce01(__builtin_nontemporal_load(p3d + ib),
                                   __builtin_nontemporal_load(y3d + ib));
        if (m4 != 0.f) a5 += bce01(__builtin_nontemporal_load(p3e + ib),
                                   __builtin_nontemporal_load(y3e + ib));

        // level-1 / level-2: tiny per-sample vectors handled by low lanes
        if (lane < 2)            a0 += bce01(l1p[i2 + lane], lane ? y1b : y1a);
        if (actCC && lane < 3)   a1 += bce01(l2pcc[i3 + lane],
                                             lane == 0 ? c0 : (lane == 1 ? c1 : c2));
        if (actUR && lane < 2)   a2 += bce01(l2pur[i2 + lane], lane ? u1 : u0);
        if (lane == 0) {
            if (actCC) a3 += 1.f;
            if (actUR) a4 += 1.f;
            a6 += m0 + m1 + m2 + m3 + m4;
        }
    }

    // ---- cross-lane reduction, finished by a real WMMA ----
    // xor-butterfly within each 16-lane half: every lane then holds its
    // half's sum for each of the 7 quantities
#pragma unroll
    for (int off = 1; off < 16; off <<= 1) {
        a0 += __shfl_xor(a0, off, 32);
        a1 += __shfl_xor(a1, off, 32);
        a2 += __shfl_xor(a2, off, 32);
        a3 += __shfl_xor(a3, off, 32);
        a4 += __shfl_xor(a4, off, 32);
        a5 += __shfl_xor(a5, off, 32);
        a6 += __shfl_xor(a6, off, 32);
    }

    // A(16x4) layout: lane L<16 supplies A[L][0],A[L][1]; lane L>=16 supplies
    // A[L-16][2],A[L-16][3].  Put half-sum of quantity q=L&15 into the first
    // slot, zero in the second.  B = ones(4x16).  Then
    //   D[q][n] = H_low[q] + 0 + H_high[q] + 0 = 32-lane total of quantity q.
    const int q = lane & 15;
    const float sel = (q == 0) ? a0 : (q == 1) ? a1 : (q == 2) ? a2 :
                      (q == 3) ? a3 : (q == 4) ? a4 : (q == 5) ? a5 :
                      (q == 6) ? a6 : 0.f;
    v2f A;  A[0] = sel; A[1] = 0.f;
    v2f Bm; Bm[0] = 1.f; Bm[1] = 1.f;
    v8f C = {};
    v8f D = __builtin_amdgcn_wmma_f32_16x16x4_f32(
        /*neg_a=*/false, A, /*neg_b=*/false, Bm,
        /*c_mod=*/(short)0, C, /*reuse_a=*/false, /*reuse_b=*/false);
    // D[k] on lanes 0..15 (C/D layout: VGPR k, lanes 0-15 => row M=k) holds total[k]

    __shared__ float sm[WAVES * 8];
    const int w = threadIdx.x >> 5;
    if (lane == 0) {
#pragma unroll
        for (int k = 0; k < 7; ++k) sm[w * 8 + k] = D[k];
    }
    __syncthreads();
    if (threadIdx.x == 0) {
#pragma unroll
        for (int k = 0; k < 7; ++k) {
            float s = 0.f;
            for (int v = 0; v < WAVES; ++v) s += sm[v * 8 + k];
            ws[blockIdx.x * 8 + k] = s;   // per-block partials, no atomics
        }
    }
}

__global__ __launch_bounds__(256) void hml_finalize(const float* __restrict__ ws,
                                                    float* __restrict__ out,
                                                    int nb, int n)
{
    __shared__ float red[256];
    const int t = threadIdx.x;
    const int k = t & 7;        // quantity (0..6 used)
    const int slice = t >> 3;   // 32 slices over blocks
    float s = 0.f;
    for (int b = slice; b < nb; b += 32) s += ws[b * 8 + k];
    red[t] = s;
    __syncthreads();
    if (t < 8) {
        float tot = 0.f;
        for (int v = 0; v < 32; ++v) tot += red[v * 8 + t];
        red[t] = tot;
    }
    __syncthreads();
    if (t == 0) {
        const float s1 = red[0], s2cc = red[1], s2ur = red[2];
        const float cCC = red[3], cUR = red[4], s3 = red[5], c3 = red[6];
        const float l1    = s1 / (2.0f * (float)n);
        const float l2cc  = cCC > 0.f ? s2cc / (cCC * 3.0f) : 0.f;
        const float l2ur  = cUR > 0.f ? s2ur / (cUR * 2.0f) : 0.f;
        const float l2    = 0.5f * (l2cc + l2ur);
        const float l3    = c3 > 0.f ? (s3 * (1.0f / 32.0f)) / c3 : 0.f;
        out[0] = l1 + l2 + l3;
    }
}

extern "C" void kernel_launch(void* const* d_in, const int* in_sizes, int n_in,
                              void* d_out, int out_size, void* d_ws, size_t ws_size,
                              hipStream_t stream)
{
    (void)n_in; (void)out_size; (void)ws_size;
    const float* l1p   = (const float*)d_in[0];
    const float* l2pcc = (const float*)d_in[1];
    const float* l2pur = (const float*)d_in[2];
    const float* p3a   = (const float*)d_in[3];
    const float* p3b   = (const float*)d_in[4];
    const float* p3c   = (const float*)d_in[5];
    const float* p3d   = (const float*)d_in[6];
    const float* p3e   = (const float*)d_in[7];
    const float* l1y   = (const float*)d_in[8];
    const float* l2ycc = (const float*)d_in[9];
    const float* l2yur = (const float*)d_in[10];
    const float* y3a   = (const float*)d_in[11];
    const float* y3b   = (const float*)d_in[12];
    const float* y3c   = (const float*)d_in[13];
    const float* y3d   = (const float*)d_in[14];
    const float* y3e   = (const float*)d_in[15];

    float* ws  = (float*)d_ws;
    float* out = (float*)d_out;
    const int n = in_sizes[0] / 2;   // B

    hml_main<<<NB, BLK, 0, stream>>>(l1p, l2pcc, l2pur, p3a, p3b, p3c, p3d, p3e,
                                     l1y, l2ycc, l2yur, y3a, y3b, y3c, y3d, y3e,
                                     ws, n);
    hml_finalize<<<1, 256, 0, stream>>>(ws, out, NB, n);
}